// KnnModule_8435315770079
// MI455X (gfx1250) — compile-verified
//
#include <hip/hip_runtime.h>
#include <stdint.h>

// ---------------------------------------------------------------------------
// kNN classifier for MI455X (gfx1250, wave32, WMMA)
//   sim = features[1024x512] @ train[200000x512]^T   (f32 WMMA 16x16x4)
//   top-200 per row via histogram radix-select + recompute-collect
//   softmax(top/T) -> prefix-k label histograms for k in {10,20,100,200}
// ---------------------------------------------------------------------------

typedef __attribute__((ext_vector_type(2))) float v2f;
typedef __attribute__((ext_vector_type(8))) float v8f;

#define D_DIM   512
#define TM      32                 // rows (features) per workgroup
#define TN      64                 // cols (train) per workgroup step
#define TK      64                 // K per LDS chunk
#define KCHUNKS (D_DIM / TK)
#define NTILES_PER_WG 25
#define NCOLS_PER_WG  (TN * NTILES_PER_WG)   // 1600
#define NBUCKET 256
#define CAP     2048
#define MAXK    200
#define NCLS    1000
#define PAD     68                 // LDS row stride in floats
#define SORT_N  2048
#define TEMP    0.07f

__device__ __forceinline__ int bucket_of(float v) {
    int b = (int)floorf(v) + 128;
    return b < 0 ? 0 : (b > 255 ? 255 : b);
}
__device__ __forceinline__ unsigned fkey(float v) {
    unsigned u = __float_as_uint(v);
    return (u & 0x80000000u) ? ~u : (u | 0x80000000u);
}
__device__ __forceinline__ float fdecode(unsigned k) {
    unsigned u = (k & 0x80000000u) ? (k ^ 0x80000000u) : ~k;
    return __uint_as_float(u);
}

// One 16x16 output tile per wave, accumulated over full K=512 through LDS
// chunks. 8 waves cover a 32x64 workgroup tile (2 M-tiles x 4 N-tiles).
__device__ __forceinline__ v8f gemm_tile(const float* __restrict__ A,
                                         const float* __restrict__ Btr,
                                         int mBase, int nBase, int Brows, int N,
                                         float* __restrict__ As,
                                         float* __restrict__ Bs, int tid) {
    const int lane = tid & 31;
    const int wave = tid >> 5;
    const int mt   = wave >> 2;    // 0..1
    const int nt   = wave & 3;     // 0..3
    const int half = lane >> 4;    // K-split select
    const int l15  = lane & 15;
    v8f acc = {};
    for (int kc = 0; kc < KCHUNKS; ++kc) {
        __syncthreads();   // previous consumers done / hist init visible
        // A tile: 32x64 floats = 512 float4, 2 per thread
        #pragma unroll
        for (int i = 0; i < 2; ++i) {
            int f4 = tid + i * 256;
            int r = f4 >> 4, c4 = f4 & 15;
            int gr = mBase + r; if (gr >= Brows) gr = Brows - 1;
            float4 v = *(const float4*)(A + (size_t)gr * D_DIM + kc * TK + c4 * 4);
            float* dst = As + r * PAD + c4 * 4;
            dst[0] = v.x; dst[1] = v.y; dst[2] = v.z; dst[3] = v.w;
        }
        // B tile: 64x64 floats = 1024 float4, 4 per thread
        #pragma unroll
        for (int i = 0; i < 4; ++i) {
            int f4 = tid + i * 256;
            int r = f4 >> 4, c4 = f4 & 15;
            int gn = nBase + r; if (gn >= N) gn = N - 1;
            float4 v = *(const float4*)(Btr + (size_t)gn * D_DIM + kc * TK + c4 * 4);
            float* dst = Bs + r * PAD + c4 * 4;
            dst[0] = v.x; dst[1] = v.y; dst[2] = v.z; dst[3] = v.w;
        }
        __syncthreads();
        const float* arow = As + (mt * 16 + l15) * PAD;
        const float* brow = Bs + (nt * 16 + l15) * PAD;
        #pragma unroll
        for (int kk = 0; kk < TK / 4; ++kk) {
            int k0 = kk * 4 + 2 * half;
            v2f a, b;
            a.x = arow[k0]; a.y = arow[k0 + 1];
            b.x = brow[k0]; b.y = brow[k0 + 1];
            acc = __builtin_amdgcn_wmma_f32_16x16x4_f32(
                false, a, false, b, (short)0, acc, false, false);
        }
    }
    return acc;
}

// Pass A: GEMM + per-row value histograms (LDS aggregate, global flush)
__global__ void __launch_bounds__(256)
knn_hist_kernel(const float* __restrict__ A, const float* __restrict__ Btr,
                unsigned* __restrict__ histG, int Brows, int N) {
    __shared__ float As[TM * PAD];
    __shared__ float Bs[TN * PAD];
    __shared__ unsigned hist[TM * NBUCKET];
    const int tid = threadIdx.x;
    const int mBase = blockIdx.x * TM;
    const int nChunk = blockIdx.y * NCOLS_PER_WG;
    for (int i = tid; i < TM * NBUCKET; i += 256) hist[i] = 0u;

    const int lane = tid & 31, wave = tid >> 5;
    const int mt = wave >> 2, nt = wave & 3, half = lane >> 4, l15 = lane & 15;

    for (int ntile = 0; ntile < NTILES_PER_WG; ++ntile) {
        int nBase = nChunk + ntile * TN;
        if (nBase >= N) break;                       // uniform across block
        v8f acc = gemm_tile(A, Btr, mBase, nBase, Brows, N, As, Bs, tid);
        int col = nBase + nt * 16 + l15;
        if (col < N) {
            #pragma unroll
            for (int v = 0; v < 8; ++v) {
                int rl = mt * 16 + v + 8 * half;
                if (mBase + rl < Brows)
                    atomicAdd(&hist[rl * NBUCKET + bucket_of(acc[v])], 1u);
            }
        }
    }
    __syncthreads();
    for (int i = tid; i < TM * NBUCKET; i += 256) {
        int rl = i >> 8, b = i & 255;
        int grow = mBase + rl;
        if (grow < Brows && hist[i])
            atomicAdd(&histG[(size_t)grow * NBUCKET + b], hist[i]);
    }
}

// Pass B: per-row threshold bucket so that count(bucket >= t) >= 200
__global__ void knn_thresh_kernel(const unsigned* __restrict__ histG,
                                  int* __restrict__ thresh, int Brows) {
    int r = blockIdx.x * blockDim.x + threadIdx.x;
    if (r >= Brows) return;
    unsigned cum = 0; int t = 0;
    for (int b = NBUCKET - 1; b >= 0; --b) {
        cum += histG[(size_t)r * NBUCKET + b];
        if (cum >= MAXK) { t = b; break; }
    }
    thresh[r] = t;
}

// Pass C: recompute GEMM, collect candidates >= per-row threshold
__global__ void __launch_bounds__(256)
knn_collect_kernel(const float* __restrict__ A, const float* __restrict__ Btr,
                   const int* __restrict__ thresh, int* __restrict__ cnt,
                   float* __restrict__ candV, int* __restrict__ candI,
                   int Brows, int N) {
    __shared__ float As[TM * PAD];
    __shared__ float Bs[TN * PAD];
    __shared__ int thr[TM];
    const int tid = threadIdx.x;
    const int mBase = blockIdx.x * TM;
    const int nChunk = blockIdx.y * NCOLS_PER_WG;
    if (tid < TM) { int g = mBase + tid; thr[tid] = (g < Brows) ? thresh[g] : NBUCKET; }

    const int lane = tid & 31, wave = tid >> 5;
    const int mt = wave >> 2, nt = wave & 3, half = lane >> 4, l15 = lane & 15;

    for (int ntile = 0; ntile < NTILES_PER_WG; ++ntile) {
        int nBase = nChunk + ntile * TN;
        if (nBase >= N) break;
        v8f acc = gemm_tile(A, Btr, mBase, nBase, Brows, N, As, Bs, tid);
        int col = nBase + nt * 16 + l15;
        if (col < N) {
            #pragma unroll
            for (int v = 0; v < 8; ++v) {
                int rl = mt * 16 + v + 8 * half;
                int grow = mBase + rl;
                if (grow < Brows && bucket_of(acc[v]) >= thr[rl]) {
                    int pos = atomicAdd(&cnt[grow], 1);
                    if (pos < CAP) {
                        candV[(size_t)grow * CAP + pos] = acc[v];
                        candI[(size_t)grow * CAP + pos] = col;
                    }
                }
            }
        }
    }
}

// Pass D: per-row bitonic sort (value desc, index asc), softmax, prefix-k hists
__global__ void __launch_bounds__(256)
knn_finalize_kernel(const float* __restrict__ candV, const int* __restrict__ candI,
                    const int* __restrict__ cnt, const int* __restrict__ labels,
                    float* __restrict__ out, int Brows) {
    __shared__ unsigned long long key[SORT_N];
    __shared__ float accS[4 * NCLS];
    __shared__ float red[256];
    const int row = blockIdx.x;
    const int tid = threadIdx.x;
    int M = cnt[row]; if (M > CAP) M = CAP;

    for (int i = tid; i < SORT_N; i += 256) {
        unsigned long long k = 0ull;
        if (i < M) {
            float v = candV[(size_t)row * CAP + i];
            unsigned idx = (unsigned)candI[(size_t)row * CAP + i];
            k = ((unsigned long long)fkey(v) << 32) | (unsigned)(~idx);
        }
        key[i] = k;
    }
    __syncthreads();
    for (int ksz = 2; ksz <= SORT_N; ksz <<= 1) {
        for (int j = ksz >> 1; j > 0; j >>= 1) {
            for (int i = tid; i < SORT_N; i += 256) {
                int ixj = i ^ j;
                if (ixj > i) {
                    bool descBlk = ((i & ksz) == 0);
                    unsigned long long a = key[i], b = key[ixj];
                    if (descBlk ? (a < b) : (a > b)) { key[i] = b; key[ixj] = a; }
                }
            }
            __syncthreads();
        }
    }
    int kTop = M < MAXK ? M : MAXK;
    float vmax = fdecode((unsigned)(key[0] >> 32));
    float e = 0.0f; int myL = 0;
    if (tid < kTop) {
        float v = fdecode((unsigned)(key[tid] >> 32));
        unsigned idx = ~((unsigned)(key[tid] & 0xFFFFFFFFu));
        myL = labels[idx];
        e = expf((v - vmax) / TEMP);
    }
    red[tid] = e;
    __syncthreads();
    for (int s = 128; s > 0; s >>= 1) {
        if (tid < s) red[tid] += red[tid + s];
        __syncthreads();
    }
    float sum = red[0];
    for (int i = tid; i < 4 * NCLS; i += 256) accS[i] = 0.0f;
    __syncthreads();
    if (tid < kTop) {
        float w = e / sum;
        const int Ks[4] = {10, 20, 100, 200};
        #pragma unroll
        for (int q = 0; q < 4; ++q)
            if (tid < Ks[q]) atomicAdd(&accS[q * NCLS + myL], w);
    }
    __syncthreads();
    for (int i = tid; i < 4 * NCLS; i += 256) {
        int q = i / NCLS, c = i - q * NCLS;
        out[(size_t)q * Brows * NCLS + (size_t)row * NCLS + c] = accS[i];
    }
}

extern "C" void kernel_launch(void* const* d_in, const int* in_sizes, int n_in,
                              void* d_out, int out_size, void* d_ws, size_t ws_size,
                              hipStream_t stream) {
    const float* A      = (const float*)d_in[0];   // [B, 512]
    const float* Btr    = (const float*)d_in[1];   // [N, 512]
    const int*   labels = (const int*)d_in[2];     // [N]
    const int B = in_sizes[0] / D_DIM;             // 1024
    const int N = in_sizes[2];                     // 200000

    char* p = (char*)d_ws;
    unsigned* histG = (unsigned*)p; p += (size_t)B * NBUCKET * sizeof(unsigned);
    int* thresh     = (int*)p;      p += (size_t)B * sizeof(int);
    int* cnt        = (int*)p;      p += (size_t)B * sizeof(int);
    float* candV    = (float*)p;    p += (size_t)B * CAP * sizeof(float);
    int* candI      = (int*)p;      p += (size_t)B * CAP * sizeof(int);

    hipMemsetAsync(histG, 0, (size_t)B * NBUCKET * sizeof(unsigned), stream);
    hipMemsetAsync(cnt, 0, (size_t)B * sizeof(int), stream);

    dim3 block(256);
    dim3 gridG((B + TM - 1) / TM, (N + NCOLS_PER_WG - 1) / NCOLS_PER_WG);
    knn_hist_kernel<<<gridG, block, 0, stream>>>(A, Btr, histG, B, N);
    knn_thresh_kernel<<<(B + 255) / 256, block, 0, stream>>>(histG, thresh, B);
    knn_collect_kernel<<<gridG, block, 0, stream>>>(A, Btr, thresh, cnt, candV, candI, B, N);
    knn_finalize_kernel<<<B, block, 0, stream>>>(candV, candI, cnt, labels, (float*)d_out, B);
}